// SelectiveSSMBlock_70763881169251
// MI455X (gfx1250) — compile-verified
//
#include <hip/hip_runtime.h>
#include <hip/hip_bf16.h>
#include <math.h>

// ---------------- problem constants ----------------
#define BATCH   2
#define SEQ     2048
#define DM      1024          // d_model
#define DI      2048          // d_inner
#define DST     16            // d_state
#define DCONV   4
#define BT      (BATCH * SEQ) // 4096 flattened rows
#define LN_EPS  1e-5f

typedef __attribute__((ext_vector_type(2))) float v2f;
typedef __attribute__((ext_vector_type(8))) float v8f;

// =====================================================================
// Kernel 1/3/5: OUT[M,N] = X[M,K] @ W[N,K]^T + bias[N]   (fp32 WMMA)
// Block = 256 threads = 8 wave32s; block tile 128x64; wave tile 32x32.
// M % 128 == 0, K % 4 == 0, N arbitrary (edge-guarded).
// =====================================================================
__global__ __launch_bounds__(256)
void gemm_xwt_wmma_f32(const float* __restrict__ X, const float* __restrict__ W,
                       const float* __restrict__ bias, float* __restrict__ OUT,
                       int M, int N, int K)
{
    const int lane  = threadIdx.x & 31;
    const int wave  = threadIdx.x >> 5;
    const int half  = lane >> 4;   // K sub-pair select (K 0,1 vs 2,3)
    const int l16   = lane & 15;

    const int waveM = wave >> 1;   // 0..3
    const int waveN = wave & 1;    // 0..1
    const int mBase = blockIdx.x * 128 + waveM * 32;
    const int nBase = blockIdx.y * 64  + waveN * 32;

    v8f acc00 = {}, acc01 = {}, acc10 = {}, acc11 = {};

    // A fragment rows (lane = M row within 16-row subtile)
    const float* xr0 = X + (size_t)(mBase +      l16) * K;
    const float* xr1 = X + (size_t)(mBase + 16 + l16) * K;
    // B fragment rows: W is (N,K) row-major, lane = N column; clamp for edges
    const int n0 = nBase +      l16;
    const int n1 = nBase + 16 + l16;
    const float* wr0 = W + (size_t)(n0 < N ? n0 : N - 1) * K;
    const float* wr1 = W + (size_t)(n1 < N ? n1 : N - 1) * K;
    const int ko = 2 * half;       // per-lane K offset within 4-wide step

    #pragma unroll 4
    for (int k = 0; k < K; k += 4) {
        v2f a0 = *(const v2f*)(xr0 + k + ko);
        v2f a1 = *(const v2f*)(xr1 + k + ko);
        v2f b0 = *(const v2f*)(wr0 + k + ko);
        v2f b1 = *(const v2f*)(wr1 + k + ko);
        acc00 = __builtin_amdgcn_wmma_f32_16x16x4_f32(false, a0, false, b0, (short)0, acc00, false, false);
        acc01 = __builtin_amdgcn_wmma_f32_16x16x4_f32(false, a0, false, b1, (short)0, acc01, false, false);
        acc10 = __builtin_amdgcn_wmma_f32_16x16x4_f32(false, a1, false, b0, (short)0, acc10, false, false);
        acc11 = __builtin_amdgcn_wmma_f32_16x16x4_f32(false, a1, false, b1, (short)0, acc11, false, false);
    }

    // C/D layout: VGPR r -> row (r + 8*half) within subtile, lane l16 -> column
    v8f accs[2][2] = { { acc00, acc01 }, { acc10, acc11 } };
    #pragma unroll
    for (int mi = 0; mi < 2; ++mi) {
        #pragma unroll
        for (int ni = 0; ni < 2; ++ni) {
            const int col = nBase + ni * 16 + l16;
            if (col >= N) continue;
            const float bv   = bias[col];
            const int   rowb = mBase + mi * 16 + 8 * half;
            #pragma unroll
            for (int r = 0; r < 8; ++r) {
                OUT[(size_t)(rowb + r) * N + col] = accs[mi][ni][r] + bv;
            }
        }
    }
}

// =====================================================================
// Kernel 2: depthwise causal conv (k=4) + bias + SiLU
// xz: (BT, 2*DI) rows; x_main = cols [0,DI). Output xc: (BT, DI).
// =====================================================================
__global__ __launch_bounds__(256)
void conv_silu_kernel(const float* __restrict__ xz, const float* __restrict__ cw,
                      const float* __restrict__ cb, float* __restrict__ xc)
{
    const int idx = blockIdx.x * 256 + threadIdx.x;     // over BT*DI
    if (idx >= BT * DI) return;
    const int c  = idx & (DI - 1);
    const int bt = idx >> 11;                            // DI == 2048
    const int t  = bt & (SEQ - 1);

    const float* w = cw + c * DCONV;                     // conv_w[c,0,:]
    float acc = cb[c];
    #pragma unroll
    for (int j = 0; j < DCONV; ++j) {
        const int tt = t - (DCONV - 1) + j;
        if (tt >= 0)
            acc += w[j] * xz[(size_t)(bt + (tt - t)) * (2 * DI) + c];
    }
    xc[idx] = acc / (1.0f + __expf(-acc));               // SiLU
}

// =====================================================================
// Kernel 4: selective scan. One lane = one (channel,state); 2 channels
// per wave (lanes 0-15 = d0 states, 16-31 = d1 states). 2048 waves.
// Reads conv output xc, writes y*silu(z) back into xc in place.
// =====================================================================
__global__ __launch_bounds__(256)
void scan_kernel(const float* __restrict__ ssm, const float* __restrict__ xz,
                 const float* __restrict__ A_log, const float* __restrict__ Dp,
                 float* __restrict__ xc)
{
    const int gtid  = blockIdx.x * 256 + threadIdx.x;
    const int wid   = gtid >> 5;                 // 0 .. BATCH*DI/2 - 1
    const int lane  = threadIdx.x & 31;
    const int half  = lane >> 4;
    const int n     = lane & 15;
    const int b     = wid >> 10;                 // wid / (DI/2)
    const int dpair = wid & 1023;
    const int d     = dpair * 2 + half;

    const float A  = -__expf(A_log[d * DST + n]);
    const float Dd = Dp[d];
    float h = 0.0f;
    const size_t btBase = (size_t)b * SEQ;

    for (int t = 0; t < SEQ; ++t) {
        const size_t bt   = btBase + t;
        const float* srow = ssm + bt * (2 * DST + 1);
        const float  Bn   = srow[n];
        const float  Cn   = srow[DST + n];
        const float  dtr  = srow[2 * DST];
        const float  dt   = (dtr > 20.0f) ? dtr : log1pf(__expf(dtr));
        const float  xv   = xc[bt * DI + d];

        h = __expf(A * dt) * h + dt * Bn * xv;

        float p = h * Cn;                        // reduce over 16 states
        p += __shfl_xor(p, 8, 32);
        p += __shfl_xor(p, 4, 32);
        p += __shfl_xor(p, 2, 32);
        p += __shfl_xor(p, 1, 32);

        const float y  = p + Dd * xv;
        const float z  = xz[bt * (2 * DI) + DI + d];
        const float yf = y * z / (1.0f + __expf(-z));   // y * silu(z)
        if (n == 0) xc[bt * DI + d] = yf;
    }
}

// =====================================================================
// Kernel 6: residual add + LayerNorm over DM=1024, one block per row
// =====================================================================
__global__ __launch_bounds__(256)
void resid_ln_kernel(const float* __restrict__ yproj, const float* __restrict__ x,
                     const float* __restrict__ gamma, const float* __restrict__ beta,
                     float* __restrict__ out)
{
    __shared__ float s1[256], s2[256];
    const int row = blockIdx.x;
    const float* yr = yproj + (size_t)row * DM;
    const float* xr = x     + (size_t)row * DM;

    float r[DM / 256];
    float sum = 0.0f, sumsq = 0.0f;
    #pragma unroll
    for (int j = 0; j < DM / 256; ++j) {
        const int i = threadIdx.x + j * 256;
        r[j] = yr[i] + xr[i];
        sum   += r[j];
        sumsq += r[j] * r[j];
    }
    s1[threadIdx.x] = sum;
    s2[threadIdx.x] = sumsq;
    __syncthreads();
    for (int off = 128; off > 0; off >>= 1) {
        if (threadIdx.x < off) {
            s1[threadIdx.x] += s1[threadIdx.x + off];
            s2[threadIdx.x] += s2[threadIdx.x + off];
        }
        __syncthreads();
    }
    const float mu  = s1[0] * (1.0f / DM);
    const float var = s2[0] * (1.0f / DM) - mu * mu;
    const float inv = rsqrtf(var + LN_EPS);
    #pragma unroll
    for (int j = 0; j < DM / 256; ++j) {
        const int i = threadIdx.x + j * 256;
        out[(size_t)row * DM + i] = (r[j] - mu) * inv * gamma[i] + beta[i];
    }
}

// =====================================================================
extern "C" void kernel_launch(void* const* d_in, const int* in_sizes, int n_in,
                              void* d_out, int out_size, void* d_ws, size_t ws_size,
                              hipStream_t stream) {
    const float* x          = (const float*)d_in[0];
    const float* in_proj_w  = (const float*)d_in[1];
    const float* in_proj_b  = (const float*)d_in[2];
    const float* conv_w     = (const float*)d_in[3];
    const float* conv_b     = (const float*)d_in[4];
    const float* x_proj_w   = (const float*)d_in[5];
    const float* x_proj_b   = (const float*)d_in[6];
    const float* A_log      = (const float*)d_in[7];
    const float* Dp         = (const float*)d_in[8];
    const float* out_proj_w = (const float*)d_in[9];
    const float* out_proj_b = (const float*)d_in[10];
    const float* ln_gamma   = (const float*)d_in[11];
    const float* ln_beta    = (const float*)d_in[12];
    float* out = (float*)d_out;

    // workspace layout (floats)
    float* ws    = (float*)d_ws;
    float* xz    = ws;                                   // BT x 4096
    float* xc    = xz    + (size_t)BT * (2 * DI);        // BT x 2048 (conv out, then y*silu(z))
    float* ssm   = xc    + (size_t)BT * DI;              // BT x 33
    float* yproj = ssm   + (size_t)BT * (2 * DST + 1);   // BT x 1024

    // 1) in_proj: xz = x @ in_proj_w^T + b      (M=4096, N=4096, K=1024)
    {
        dim3 grid(BT / 128, (2 * DI + 63) / 64);
        gemm_xwt_wmma_f32<<<grid, 256, 0, stream>>>(x, in_proj_w, in_proj_b, xz,
                                                    BT, 2 * DI, DM);
    }
    // 2) depthwise conv + SiLU: xc
    conv_silu_kernel<<<(BT * DI) / 256, 256, 0, stream>>>(xz, conv_w, conv_b, xc);

    // 3) x_proj: ssm = xc @ x_proj_w^T + b      (M=4096, N=33, K=2048)
    {
        dim3 grid(BT / 128, (2 * DST + 1 + 63) / 64);
        gemm_xwt_wmma_f32<<<grid, 256, 0, stream>>>(xc, x_proj_w, x_proj_b, ssm,
                                                    BT, 2 * DST + 1, DI);
    }
    // 4) selective scan, writes y*silu(z) in place into xc
    scan_kernel<<<(BATCH * (DI / 2) * 32) / 256, 256, 0, stream>>>(ssm, xz, A_log, Dp, xc);

    // 5) out_proj: yproj = xc @ out_proj_w^T + b (M=4096, N=1024, K=2048)
    {
        dim3 grid(BT / 128, (DM + 63) / 64);
        gemm_xwt_wmma_f32<<<grid, 256, 0, stream>>>(xc, out_proj_w, out_proj_b, yproj,
                                                    BT, DM, DI);
    }
    // 6) residual + LayerNorm
    resid_ln_kernel<<<BT, 256, 0, stream>>>(yproj, x, ln_gamma, ln_beta, out);
}